// MultiScaleGraphNetwork_31963146617559
// MI455X (gfx1250) — compile-verified
//
#include <hip/hip_runtime.h>

typedef __attribute__((ext_vector_type(2))) float v2f;
typedef __attribute__((ext_vector_type(8))) float v8f;

// Hardware FP32 atomic add (global_atomic_add_f32) — the L2 does the RMW.
__device__ __forceinline__ void atomic_add_f32(float* p, float v) {
    __hip_atomic_fetch_add(p, v, __ATOMIC_RELAXED, __HIP_MEMORY_SCOPE_AGENT);
}

// ---------------- degree / norm ----------------
__global__ void deg_kernel(const int* __restrict__ dst, int E, int N,
                           float* __restrict__ deg) {
    int i = blockIdx.x * blockDim.x + threadIdx.x;
    int total = E + N;
    if (i >= total) return;
    int d = (i < E) ? dst[i] : (i - E);   // self-loops for i >= E
    atomic_add_f32(&deg[d], 1.0f);
}

__global__ void rsqrt_kernel(float* __restrict__ dis, int N) {
    int i = blockIdx.x * blockDim.x + threadIdx.x;
    if (i >= N) return;
    dis[i] = rsqrtf(fmaxf(dis[i], 1.0f));
}

// norm[e] = dis[src]*dis[dst]; self-loop tail: norm[E+n] = dis[n]^2
__global__ void norm_kernel(const int* __restrict__ src, const int* __restrict__ dstv,
                            int E, int N, const float* __restrict__ dis,
                            float* __restrict__ norm) {
    int e = blockIdx.x * blockDim.x + threadIdx.x;
    if (e >= E + N) return;
    int s, d;
    if (e < E) { s = src[e]; d = dstv[e]; } else { s = d = e - E; }
    norm[e] = dis[s] * dis[d];
}

// ---------------- fp32 WMMA GEMM: out[:, outcol0:outcol0+96] = A[K] @ W[Kx96] (+bias) ----
// One wave32 computes one 16x16 output tile via V_WMMA_F32_16X16X4_F32.
__global__ __launch_bounds__(256)
void gemm_wmma_f32(const float* __restrict__ A, int lda, int K,
                   const float* __restrict__ W,
                   float* __restrict__ out, int ldo, int outcol0,
                   const float* __restrict__ bias, int row_tiles) {
    int wave  = blockIdx.x * (blockDim.x >> 5) + (threadIdx.x >> 5);
    int total = row_tiles * 6;                 // 96 output cols = 6 tiles
    if (wave >= total) return;                 // wave-uniform exit: EXEC all-1s for WMMA

    int row_tile = wave / 6;
    int col_tile = wave % 6;
    int lane = threadIdx.x & 31;
    int m    = lane & 15;                      // row-in-tile for A, col-in-tile for B/D
    int half = lane >> 4;                      // K sub-slot select

    const float* arow = A + (size_t)(row_tile * 16 + m) * lda;
    int wcol = col_tile * 16 + m;

    v8f acc = {};
    for (int k = 0; k < K; k += 4) {
        // A 16x4 f32 fragment: lane(0-15) holds K=k,k+1 ; lane(16-31) holds K=k+2,k+3
        v2f a = *(const v2f*)(arow + k + 2 * half);
        // B 4x16 f32 fragment: V0 = row K=k (+2*half), V1 = row K=k+1 (+2*half)
        v2f b;
        b.x = W[(size_t)(k + 2 * half)     * 96 + wcol];
        b.y = W[(size_t)(k + 2 * half + 1) * 96 + wcol];
        acc = __builtin_amdgcn_wmma_f32_16x16x4_f32(
                  false, a, false, b, (short)0, acc, false, false);
    }

    int   gcol = outcol0 + wcol;
    float bv   = bias ? bias[gcol] : 0.0f;
    // D 16x16 f32: VGPR r -> M = r + 8*half, N = lane&15
    #pragma unroll
    for (int r = 0; r < 8; ++r) {
        int row = row_tile * 16 + r + 8 * half;
        out[(size_t)row * ldo + gcol] = acc[r] + bv;
    }
}

// ---------------- edge scatter: out[dst] += norm * h[src], float4 quads ----------------
// One thread handles one (edge, 4-feature quad): b128 gather + 4 f32 atomics.
__global__ void agg_kernel(const int* __restrict__ src, const int* __restrict__ dstv,
                           int E, int N, int Q,                 // Q = F/4 quads per edge
                           const float* __restrict__ norm,
                           const float* __restrict__ h, int ldh,
                           float* __restrict__ out, int ldo, int ocol0) {
    long long gid   = (long long)blockIdx.x * blockDim.x + threadIdx.x;
    long long total = (long long)(E + N) * Q;
    if (gid >= total) return;
    int e = (int)(gid / Q);
    int q = (int)(gid % Q);
    int s, d;
    if (e < E) { s = src[e]; d = dstv[e]; } else { s = d = e - E; }
    float nrm = norm[e];
    const float4 hv = *(const float4*)(h + (size_t)s * ldh + q * 4);
    float* o = out + (size_t)d * ldo + ocol0 + q * 4;
    atomic_add_f32(o + 0, nrm * hv.x);
    atomic_add_f32(o + 1, nrm * hv.y);
    atomic_add_f32(o + 2, nrm * hv.z);
    atomic_add_f32(o + 3, nrm * hv.w);
}

// relu(in-place) over agg_cat[N x 192] with two 96-wide biases
__global__ void relu_bias2(float* __restrict__ io,
                           const float* __restrict__ b0,
                           const float* __restrict__ b1, int N) {
    long long i = (long long)blockIdx.x * blockDim.x + threadIdx.x;
    if (i >= (long long)N * 192) return;
    int f = (int)(i % 192);
    float c = (f < 96) ? b0[f] : b1[f - 96];
    io[i] = fmaxf(io[i] + c, 0.0f);
}

// out[n, ocol0+f] = relu(in[n, icol0+f] + b[f])
__global__ void relu_move(const float* __restrict__ in, int ldi, int icol0,
                          const float* __restrict__ b,
                          float* __restrict__ out, int ldo, int ocol0,
                          int N, int F) {
    long long i = (long long)blockIdx.x * blockDim.x + threadIdx.x;
    if (i >= (long long)N * F) return;
    int n = (int)(i / F), f = (int)(i % F);
    out[(size_t)n * ldo + ocol0 + f] = fmaxf(in[(size_t)n * ldi + icol0 + f] + b[f], 0.0f);
}

// zero a strided column block
__global__ void zero_strided(float* __restrict__ p, int ld, int col0, int N, int F) {
    long long i = (long long)blockIdx.x * blockDim.x + threadIdx.x;
    if (i >= (long long)N * F) return;
    int n = (int)(i / F), f = (int)(i % F);
    p[(size_t)n * ld + col0 + f] = 0.0f;
}

static inline int cdiv(long long a, int b) { return (int)((a + b - 1) / b); }

extern "C" void kernel_launch(void* const* d_in, const int* in_sizes, int n_in,
                              void* d_out, int out_size, void* d_ws, size_t ws_size,
                              hipStream_t stream) {
    const float* x       = (const float*)d_in[0];
    const int*   ei      = (const int*)  d_in[1];
    const float* W_local = (const float*)d_in[2];
    const float* b_local = (const float*)d_in[3];
    const float* W_g1    = (const float*)d_in[4];
    const float* b_g1    = (const float*)d_in[5];
    const float* W_g2    = (const float*)d_in[6];
    const float* b_g2    = (const float*)d_in[7];
    const float* W_fuse  = (const float*)d_in[8];
    const float* b_fuse  = (const float*)d_in[9];

    const int N = in_sizes[0] / 96;
    const int E = in_sizes[1] / 2;
    const int* src = ei;
    const int* dst = ei + E;

    float* ws    = (float*)d_ws;
    float* dis   = ws;                         // [N]
    float* norm  = ws + N;                     // [E + N]
    float* h_cat = norm + (E + N);             // [N x 192]  (GEMM outputs / layer-2 scratch)
    float* agg   = h_cat + (size_t)N * 192;    // [N x 192]  (aggregates -> combined)

    hipMemsetAsync(dis, 0, (size_t)N * sizeof(float), stream);
    hipMemsetAsync(agg, 0, (size_t)N * 192 * sizeof(float), stream);

    const int B = 256;

    // degrees + symmetric normalization, then per-edge norm weights
    deg_kernel  <<<cdiv(E + N, B), B, 0, stream>>>(dst, E, N, dis);
    rsqrt_kernel<<<cdiv(N, B),     B, 0, stream>>>(dis, N);
    norm_kernel <<<cdiv(E + N, B), B, 0, stream>>>(src, dst, E, N, dis, norm);

    const int row_tiles = N / 16;              // N = 50000 = 3125 * 16, exact
    const int gblocks   = cdiv((long long)row_tiles * 6, 8);  // 8 waves per block

    // layer 1: h_cat[:,0:96] = x @ W_local ; h_cat[:,96:192] = x @ W_g1
    gemm_wmma_f32<<<gblocks, B, 0, stream>>>(x, 96, 96, W_local, h_cat, 192, 0,  nullptr, row_tiles);
    gemm_wmma_f32<<<gblocks, B, 0, stream>>>(x, 96, 96, W_g1,    h_cat, 192, 96, nullptr, row_tiles);

    // fused scatter of both 96-wide message streams (192 feats/edge = 48 quads)
    agg_kernel<<<cdiv((long long)(E + N) * 48, B), B, 0, stream>>>(
        src, dst, E, N, 48, norm, h_cat, 192, agg, 192, 0);

    // agg[:,0:96] -> local (final), agg[:,96:192] -> g
    relu_bias2<<<cdiv((long long)N * 192, B), B, 0, stream>>>(agg, b_local, b_g1, N);

    // layer 2: h_g2 = g @ W_g2 into h_cat[:,0:96]; aggregate into h_cat[:,96:192]
    zero_strided<<<cdiv((long long)N * 96, B), B, 0, stream>>>(h_cat, 192, 96, N, 96);
    gemm_wmma_f32<<<gblocks, B, 0, stream>>>(agg + 96, 192, 96, W_g2, h_cat, 192, 0, nullptr, row_tiles);
    agg_kernel<<<cdiv((long long)(E + N) * 24, B), B, 0, stream>>>(
        src, dst, E, N, 24, norm, h_cat, 192, h_cat, 192, 96);

    // g2 = relu(agg_g2 + b_g2) into agg[:,96:192]  => agg == combined [N x 192]
    relu_move<<<cdiv((long long)N * 96, B), B, 0, stream>>>(
        h_cat, 192, 96, b_g2, agg, 192, 96, N, 96);

    // fuse: out = combined @ W_fuse + b_fuse   (K = 192)
    gemm_wmma_f32<<<gblocks, B, 0, stream>>>(
        agg, 192, 192, W_fuse, (float*)d_out, 96, 0, b_fuse, row_tiles);
}